// VGG_32753420599454
// MI455X (gfx1250) — compile-verified
//
#include <hip/hip_runtime.h>

#define GROUPS 10
#define INPUTS 6
#define COLS   60          // GROUPS * INPUTS
#define EPS    1e-4f

typedef float v2f __attribute__((ext_vector_type(2)));
typedef float v4f __attribute__((ext_vector_type(4)));
typedef float v8f __attribute__((ext_vector_type(8)));

// ---------------------------------------------------------------------------
// Kernel 0: zero the 20 stats accumulators in workspace.
// ---------------------------------------------------------------------------
__global__ void init_ws_kernel(float* ws) {
    int t = threadIdx.x;
    if (t < 2 * GROUPS) ws[t] = 0.0f;
}

// ---------------------------------------------------------------------------
// Kernel 1: y = X * blockdiag(W) + b via V_WMMA_F32_16X16X4_F32.
// Each wave processes 16-row tiles. K = 60 -> 15 accumulating WMMAs.
// X is streamed with non-temporal loads so Y (41.9 MB) can stay resident in
// the 192 MB L2 for the normalize pass. Writes y to Y and accumulates
// per-group sum / sum-of-squares into ws[0..19].
// ---------------------------------------------------------------------------
__global__ void __launch_bounds__(256)
fused_gemv_stats_kernel(const float* __restrict__ X,
                        const float* __restrict__ W,
                        const float* __restrict__ bias,
                        float* __restrict__ Y,
                        float* __restrict__ ws,
                        int numTiles)
{
    const int lane   = threadIdx.x & 31;
    const int wavesPerBlk = blockDim.x >> 5;
    const int wave   = blockIdx.x * wavesPerBlk + (threadIdx.x >> 5);
    const int nWaves = gridDim.x * wavesPerBlk;
    const int n      = lane & 15;            // column index (group id for n<10)
    const int khalf  = (lane >> 4) << 1;     // 0 for lanes 0-15, 2 for 16-31
    const bool active = (n < GROUPS);

    // Precompute wave-constant B fragments for the 15 K-steps.
    // Wbig[k][n] = (k/6 == n) ? W[k] : 0   (W flat [10,6], W[g*6+i])
    v2f bfrag[15];
#pragma unroll
    for (int kk = 0; kk < 15; ++kk) {
        int k0 = 4 * kk + khalf;
        int k1 = k0 + 1;
        float w0 = W[k0];
        float w1 = W[k1];
        bfrag[kk].x = ((k0 / INPUTS) == n) ? w0 : 0.0f;
        bfrag[kk].y = ((k1 / INPUTS) == n) ? w1 : 0.0f;
    }
    const float bn = active ? bias[n] : 0.0f;

    float lsum = 0.0f, lsq = 0.0f;

    for (int tile = wave; tile < numTiles; tile += nWaves) {
        const int row0 = tile * 16;
        // A fragment: lane holds row (row0 + n), 2 consecutive K values
        // at column offset 4*kk + khalf  (8-byte aligned, non-temporal).
        const v2f* xrow = (const v2f*)(X + (size_t)(row0 + n) * COLS + khalf);

        v8f c = {};
#pragma unroll
        for (int kk = 0; kk < 15; ++kk) {
            v2f a = __builtin_nontemporal_load(xrow + 2 * kk);
            c = __builtin_amdgcn_wmma_f32_16x16x4_f32(
                    /*neg_a=*/false, a, /*neg_b=*/false, bfrag[kk],
                    /*c_mod=*/(short)0, c, /*reuse_a=*/false, /*reuse_b=*/false);
        }

        // C/D layout: lane<16 -> VGPR j = y[row0+j][n]; lane>=16 -> y[row0+8+j][n]
        // Single predicated block: one EXEC save/restore instead of eight.
        if (active) {
            const int rbase = row0 + ((lane >> 4) << 3);
            float* yp = Y + (size_t)rbase * GROUPS + n;
#pragma unroll
            for (int j = 0; j < 8; ++j) {
                float y = c[j] + bn;
                yp[(size_t)j * GROUPS] = y;      // regular (temporal) -> stays in L2
                lsum += y;
                lsq  += y * y;
            }
        }
    }

    // Fold the two half-wave partials for the same group together.
    lsum += __shfl_xor(lsum, 16, 32);
    lsq  += __shfl_xor(lsq,  16, 32);

    __shared__ float s_sum[GROUPS];
    __shared__ float s_sq[GROUPS];
    if (threadIdx.x < GROUPS) { s_sum[threadIdx.x] = 0.0f; s_sq[threadIdx.x] = 0.0f; }
    __syncthreads();
    if (lane < GROUPS) {                 // lanes 0-9 of every wave
        atomicAdd(&s_sum[n], lsum);
        atomicAdd(&s_sq[n],  lsq);
    }
    __syncthreads();
    if (threadIdx.x < GROUPS) {
        atomicAdd(&ws[threadIdx.x],          s_sum[threadIdx.x]);
        atomicAdd(&ws[GROUPS + threadIdx.x], s_sq[threadIdx.x]);
    }
}

// ---------------------------------------------------------------------------
// Kernel 2: turn (sum, sumsq) into per-group (scale, shift) in ws[20..39].
// ---------------------------------------------------------------------------
__global__ void finalize_stats_kernel(const float* __restrict__ gamma,
                                      const float* __restrict__ beta,
                                      float* ws, float invB)
{
    int g = threadIdx.x;
    if (g < GROUPS) {
        float mean = ws[g] * invB;
        float var  = ws[GROUPS + g] * invB - mean * mean;
        float sc   = gamma[g] * rsqrtf(var + EPS);
        ws[2 * GROUPS + g] = sc;
        ws[3 * GROUPS + g] = beta[g] - mean * sc;
    }
}

// ---------------------------------------------------------------------------
// Kernel 3: streaming normalize, 16-byte vectorized (B*10 divisible by 4).
// Reads should hit L2 (Y resident); final writes are non-temporal.
// ---------------------------------------------------------------------------
__global__ void __launch_bounds__(256)
normalize_kernel(float* __restrict__ Y, const float* __restrict__ ws, int n4)
{
    int i = blockIdx.x * blockDim.x + threadIdx.x;
    if (i >= n4) return;
    const float* sc = ws + 2 * GROUPS;
    const float* sh = ws + 3 * GROUPS;

    v4f* Y4 = (v4f*)Y;
    v4f v = Y4[i];
    int g = (i * 4) % GROUPS;
    v.x = v.x * sc[g] + sh[g]; g = (g == GROUPS - 1) ? 0 : g + 1;
    v.y = v.y * sc[g] + sh[g]; g = (g == GROUPS - 1) ? 0 : g + 1;
    v.z = v.z * sc[g] + sh[g]; g = (g == GROUPS - 1) ? 0 : g + 1;
    v.w = v.w * sc[g] + sh[g];
    __builtin_nontemporal_store(v, Y4 + i);
}

// ---------------------------------------------------------------------------
extern "C" void kernel_launch(void* const* d_in, const int* in_sizes, int n_in,
                              void* d_out, int out_size, void* d_ws, size_t ws_size,
                              hipStream_t stream)
{
    const float* X     = (const float*)d_in[0];   // [B, 60]
    const float* W     = (const float*)d_in[1];   // [10, 6]
    const float* bias  = (const float*)d_in[2];   // [10]
    const float* gamma = (const float*)d_in[3];   // [10]
    const float* beta  = (const float*)d_in[4];   // [10]
    float* Y  = (float*)d_out;                    // [B, 10]
    float* ws = (float*)d_ws;                     // 40 floats used

    const int B        = in_sizes[0] / COLS;      // 1048576
    const int numTiles = B / 16;                  // 65536 (B divisible by 16)

    init_ws_kernel<<<1, 32, 0, stream>>>(ws);

    // 2048 blocks x 8 waves = 16384 waves, 4 tiles each.
    fused_gemv_stats_kernel<<<2048, 256, 0, stream>>>(X, W, bias, Y, ws, numTiles);

    finalize_stats_kernel<<<1, 32, 0, stream>>>(gamma, beta, ws, 1.0f / (float)B);

    const int n4 = (B * GROUPS) / 4;              // 2,621,440
    normalize_kernel<<<(n4 + 255) / 256, 256, 0, stream>>>(Y, ws, n4);
}